// ResampleFromUVTexture_56410100465683
// MI455X (gfx1250) — compile-verified
//
#include <hip/hip_runtime.h>
#include <hip/hip_bf16.h>
#include <stdint.h>

// Bilinear resample from UV texture, MI455X (gfx1250).
// Memory-bound gather: x (84MB) is L2-resident; quad/uv streamed via CDNA5
// async global->LDS; output streamed with non-temporal stores.

namespace {

constexpr int kB  = 2;
constexpr int kC  = 8;
constexpr int kF  = 80;
constexpr int kN  = 128;
constexpr int kH  = 512;
constexpr int kW  = 1024;

constexpr int kBC  = kB * kC;    // 16 channel planes
constexpr int kHW  = kH * kW;    // 524288 pixels
constexpr int kNN  = kN * kN;    // 16384 texels per face
constexpr int kFNN = kF * kNN;   // 1310720 texels per (b,c) plane

constexpr int kThreads = 256;    // 8 wave32 waves per block

__global__ __launch_bounds__(kThreads)
void resample_bilinear_kernel(const float* __restrict__ x,
                              const int*   __restrict__ quad,
                              const float* __restrict__ uv,
                              float*       __restrict__ out)
{
    __shared__ float uv_s[kThreads * 2];   // 2 KB
    __shared__ int   quad_s[kThreads];     // 1 KB

    const int tid = threadIdx.x;
    const int pix = blockIdx.x * kThreads + tid;   // grid covers kHW exactly

    // ---- CDNA5 async global->LDS staging of the coalesced uv/quad streams.
    // Each lane copies its own pixel's uv (8B) and quad (4B) into LDS via the
    // async data path (tracked by ASYNCcnt), then waits on its own counter.
    // Low 32 bits of a generic pointer to __shared__ are the LDS byte address.
    {
        uint64_t g_uv = (uint64_t)(uintptr_t)(uv + (size_t)pix * 2);
        uint32_t l_uv = (uint32_t)(uintptr_t)(&uv_s[tid * 2]);
        asm volatile("global_load_async_to_lds_b64 %0, %1, off"
                     :: "v"(l_uv), "v"(g_uv) : "memory");

        uint64_t g_q = (uint64_t)(uintptr_t)(quad + pix);
        uint32_t l_q = (uint32_t)(uintptr_t)(&quad_s[tid]);
        asm volatile("global_load_async_to_lds_b32 %0, %1, off"
                     :: "v"(l_q), "v"(g_q) : "memory");

        asm volatile("s_wait_asynccnt 0" ::: "memory");
        // Each lane consumes only its own staged slot; the per-wave
        // ASYNCcnt wait is sufficient -- no workgroup barrier needed.
    }

    const float u = uv_s[tid * 2 + 0];
    const float v = uv_s[tid * 2 + 1];
    const int   q = quad_s[tid];

    // u0 = clip(floor(u), 0, N-2); du = u - u0  (same for v)
    float u0f = floorf(u);
    u0f = fminf(fmaxf(u0f, 0.0f), (float)(kN - 2));
    float v0f = floorf(v);
    v0f = fminf(fmaxf(v0f, 0.0f), (float)(kN - 2));

    const int   u0 = (int)u0f;
    const int   v0 = (int)v0f;
    const float du = u - u0f;
    const float dv = v - v0f;

    const float w00 = (1.0f - du) * (1.0f - dv);
    const float w01 = du * (1.0f - dv);
    const float w10 = (1.0f - du) * dv;
    const float w11 = du * dv;

    const int i00 = q * kNN + v0 * kN + u0;   // < 1.31M, fits int32

    // 16 channel planes: 4 L2-resident gathers + 3 FMAs + 1 NT store each.
    // Unrolled so ~64 gather loads are in flight per thread.
#pragma unroll
    for (int bc = 0; bc < kBC; ++bc) {
        const float* p = x + (size_t)bc * kFNN + i00;
        const float g00 = p[0];
        const float g01 = p[1];
        const float g10 = p[kN];
        const float g11 = p[kN + 1];

        const float r = fmaf(g00, w00,
                        fmaf(g01, w01,
                        fmaf(g10, w10, g11 * w11)));

        // Output is write-once streaming data: keep it out of L2 so the
        // 84MB texture stays resident (NT store -> th:TH_STORE_NT).
        __builtin_nontemporal_store(r, out + (size_t)bc * kHW + pix);
    }
}

} // namespace

extern "C" void kernel_launch(void* const* d_in, const int* /*in_sizes*/, int /*n_in*/,
                              void* d_out, int /*out_size*/, void* /*d_ws*/, size_t /*ws_size*/,
                              hipStream_t stream)
{
    const float* x    = (const float*)d_in[0];   // B x C x F x N x N  f32
    const int*   quad = (const int*)  d_in[1];   // H x W              i32
    const float* uv   = (const float*)d_in[2];   // H x W x 2          f32
    float*       out  = (float*)d_out;           // B x C x H x W      f32

    static_assert(kHW % kThreads == 0, "grid must tile pixel count exactly");
    dim3 grid(kHW / kThreads);   // 2048 blocks x 256 threads, 1 pixel/thread
    resample_bilinear_kernel<<<grid, kThreads, 0, stream>>>(x, quad, uv, out);
}